// GAN_63041529971276
// MI455X (gfx1250) — compile-verified
//
#include <hip/hip_runtime.h>
#include <hip/hip_bf16.h>

typedef __attribute__((ext_vector_type(2))) float v2f;
typedef __attribute__((ext_vector_type(8))) float v8f;
typedef int v4i_vec __attribute__((vector_size(16)));   // int __vector(4), per hipcc diagnostic

#define AS1 __attribute__((address_space(1)))
#define AS3 __attribute__((address_space(3)))
typedef AS1 v4i_vec* gptr128_t;   // global int4*
typedef AS3 v4i_vec* lptr128_t;   // LDS int4*

#define DD 256
#define LDA (DD + 4)     // pad 4 floats: lanes 0-15 -> banks 4m+k, lanes 16-31 -> 4m+k+2 (disjoint)
#define RT  4            // row tiles per block (64 rows), B fragment reused 4x

// ---------------- zero fill ----------------
__global__ void fill_zero_kernel(float* __restrict__ p, int n) {
    int i = blockIdx.x * blockDim.x + threadIdx.x;
    if (i < n) p[i] = 0.0f;
}

// ---------------- edge gather + atomic scatter-sum ----------------
// 64 threads per edge, each handles one float4 chunk of the 256-wide row.
__global__ __launch_bounds__(256) void scatter_kernel(
    const float* __restrict__ x, const int* __restrict__ edge_index,
    const int* __restrict__ edge_attr, const float* __restrict__ emb1,
    const float* __restrict__ emb2, float* __restrict__ agg, int E)
{
    long tid = (long)blockIdx.x * blockDim.x + threadIdx.x;
    long total = (long)E * 64;
    if (tid >= total) return;
    int e  = (int)(tid >> 6);
    int c4 = ((int)tid & 63) * 4;
    int src = edge_index[e];
    int dst = edge_index[E + e];
    int a0  = edge_attr[2 * e];
    int a1  = edge_attr[2 * e + 1];
    const float4 xs = *(const float4*)(x    + (long)src * DD + c4);
    const float4 e1 = *(const float4*)(emb1 + (long)a0  * DD + c4);
    const float4 e2 = *(const float4*)(emb2 + (long)a1  * DD + c4);
    float* p = agg + (long)dst * DD + c4;
    unsafeAtomicAdd(p + 0, xs.x + e1.x + e2.x);
    unsafeAtomicAdd(p + 1, xs.y + e1.y + e2.y);
    unsafeAtomicAdd(p + 2, xs.z + e1.z + e2.z);
    unsafeAtomicAdd(p + 3, xs.w + e1.w + e2.w);
}

// ---------------- L2 normalize + add noise (one wave32 per node) ----------------
__global__ __launch_bounds__(256) void finalize_kernel(
    const float* __restrict__ agg, const float* __restrict__ noise,
    float* __restrict__ h0, int n)
{
    int wave = threadIdx.x >> 5;
    int lane = threadIdx.x & 31;
    int node = blockIdx.x * 8 + wave;
    if (node >= n) return;
    const float* row = agg + (long)node * DD;
    float4 v0 = *(const float4*)(row + lane * 8);
    float4 v1 = *(const float4*)(row + lane * 8 + 4);
    float ss = v0.x * v0.x + v0.y * v0.y + v0.z * v0.z + v0.w * v0.w +
               v1.x * v1.x + v1.y * v1.y + v1.z * v1.z + v1.w * v1.w;
    #pragma unroll
    for (int off = 16; off >= 1; off >>= 1) ss += __shfl_xor(ss, off, 32);
    float inv = 1.0f / fmaxf(sqrtf(ss), 1e-12f);
    const float* nz = noise + (long)node * DD;
    float4 n0 = *(const float4*)(nz + lane * 8);
    float4 n1 = *(const float4*)(nz + lane * 8 + 4);
    float* o = h0 + (long)node * DD;
    float4 r0 = { v0.x * inv + n0.x, v0.y * inv + n0.y, v0.z * inv + n0.z, v0.w * inv + n0.w };
    float4 r1 = { v1.x * inv + n1.x, v1.y * inv + n1.y, v1.z * inv + n1.z, v1.w * inv + n1.w };
    *(float4*)(o + lane * 8)     = r0;
    *(float4*)(o + lane * 8 + 4) = r1;
}

// ---------------- fused GEMM: out = relu(A @ W^T + bias) via V_WMMA_F32_16X16X4_F32 ----
// Block = 512 threads = 16 waves; block owns 64 rows (4 row-tiles), wave w owns cols
// [16w,16w+16) for all 4 row-tiles -> each B fragment feeds 4 WMMAs (4x less W traffic).
// A tile staged in LDS via GLOBAL_LOAD_ASYNC_TO_LDS_B128 when the toolchain exposes it.
__global__ __launch_bounds__(512) void gemm_relu_kernel(
    const float* __restrict__ A, const float* __restrict__ W,
    const float* __restrict__ bias, float* __restrict__ out, int rows)
{
    extern __shared__ float As[];                  // (16*RT) x LDA floats = 66.5 KB
    const int row0 = blockIdx.x * (16 * RT);
    const int tid  = threadIdx.x;

#if defined(__gfx1250__) && __has_builtin(__builtin_amdgcn_global_load_async_to_lds_b128)
    {
        const int nvalid = rows - row0;            // >= 16*RT for all but the tail block
        // 16*RT rows x 64 float4 chunks; 512 threads -> 8 async copies per thread
        for (int i = tid; i < 16 * RT * (DD / 4); i += 512) {
            int r  = i >> 6;
            int c4 = (i & 63) * 4;
            if (r < nvalid) {
                __builtin_amdgcn_global_load_async_to_lds_b128(
                    (gptr128_t)(A + (long)(row0 + r) * DD + c4),
                    (lptr128_t)(As + r * LDA + c4),
                    /*imm_offset=*/0, /*cpol=*/0);
            } else {
                float4 z = {0.0f, 0.0f, 0.0f, 0.0f};
                *(float4*)(As + r * LDA + c4) = z;
            }
        }
#if __has_builtin(__builtin_amdgcn_s_wait_asynccnt)
        __builtin_amdgcn_s_wait_asynccnt(0);
#else
        asm volatile("s_wait_asynccnt 0x0" ::: "memory");
#endif
    }
#else
    for (int i = tid; i < 16 * RT * DD; i += 512) {
        int r = i >> 8, c = i & 255;
        int gr = row0 + r;
        As[r * LDA + c] = (gr < rows) ? A[(long)gr * DD + c] : 0.0f;
    }
#endif
    __syncthreads();

    const int wave  = tid >> 5;
    const int lane  = tid & 31;
    const int col0  = wave * 16;
    const int m     = lane & 15;
    const int khalf = lane >> 4;                       // 0 or 1
    const float* wrow = W + (long)(col0 + m) * DD;     // row of W = column of B

    v8f acc[RT] = {};
    #pragma unroll 4
    for (int k0 = 0; k0 < DD; k0 += 4) {
        const int kk = k0 + 2 * khalf;
        v2f b;
        b.x = wrow[kk];
        b.y = wrow[kk + 1];
        #pragma unroll
        for (int t = 0; t < RT; ++t) {
            v2f a;
            a.x = As[(16 * t + m) * LDA + kk];
            a.y = As[(16 * t + m) * LDA + kk + 1];
            acc[t] = __builtin_amdgcn_wmma_f32_16x16x4_f32(
                         /*neg_a=*/false, a, /*neg_b=*/false, b,
                         /*c_mod=*/(short)0, acc[t], /*reuse_a=*/false, /*reuse_b=*/false);
        }
    }

    const float bv = bias[col0 + m];
    if (row0 + 16 * RT <= rows) {
        // fast path: full tile, no per-element exec games
        #pragma unroll
        for (int t = 0; t < RT; ++t) {
            #pragma unroll
            for (int v = 0; v < 8; ++v) {
                int r = row0 + 16 * t + v + 8 * khalf;
                float val = acc[t][v] + bv;
                out[(long)r * DD + col0 + m] = fmaxf(val, 0.0f);
            }
        }
    } else {
        #pragma unroll
        for (int t = 0; t < RT; ++t) {
            #pragma unroll
            for (int v = 0; v < 8; ++v) {
                int r = row0 + 16 * t + v + 8 * khalf;
                if (r < rows) {
                    float val = acc[t][v] + bv;
                    out[(long)r * DD + col0 + m] = fmaxf(val, 0.0f);
                }
            }
        }
    }
}

extern "C" void kernel_launch(void* const* d_in, const int* in_sizes, int n_in,
                              void* d_out, int out_size, void* d_ws, size_t ws_size,
                              hipStream_t stream) {
    const float* x          = (const float*)d_in[0];
    const int*   edge_index = (const int*)d_in[1];
    const int*   edge_attr  = (const int*)d_in[2];
    // d_in[3] = batch (unused by reference)
    const float* W1   = (const float*)d_in[4];
    const float* b1   = (const float*)d_in[5];
    const float* W2   = (const float*)d_in[6];
    const float* b2   = (const float*)d_in[7];
    const float* emb1 = (const float*)d_in[8];
    const float* emb2 = (const float*)d_in[9];
    const float* noise= (const float*)d_in[10];

    const int N = in_sizes[0] / DD;
    const int E = in_sizes[1] / 2;

    float* agg = (float*)d_ws;
    float* h0  = agg + (size_t)N * DD;
    float* h1  = h0  + (size_t)N * DD;
    float* out = (float*)d_out;

    // 1) zero accumulator
    fill_zero_kernel<<<(N * DD + 255) / 256, 256, 0, stream>>>(agg, N * DD);

    // 2) gather + scatter-add (L2-resident atomics)
    long total = (long)E * 64;
    scatter_kernel<<<(int)((total + 255) / 256), 256, 0, stream>>>(
        x, edge_index, edge_attr, emb1, emb2, agg, E);

    // 3) L2 normalize + noise
    finalize_kernel<<<(N + 7) / 8, 256, 0, stream>>>(agg, noise, h0, N);

    // 4) two fused GEMM+bias+ReLU layers on the fp32 WMMA path
    const int    rowsPerBlk = 16 * RT;
    const int    blks       = (N + rowsPerBlk - 1) / rowsPerBlk;
    const size_t smem       = (size_t)(16 * RT) * LDA * sizeof(float);
    gemm_relu_kernel<<<blks, 512, smem, stream>>>(h0, W1, b1, h1, N);
    gemm_relu_kernel<<<blks, 512, smem, stream>>>(h1, W2, b2, out, N);
}